// RelationshipSummarizer_74036646248794
// MI455X (gfx1250) — compile-verified
//
#include <hip/hip_runtime.h>

#define H 128
#define NUSER 200000

typedef float v2f __attribute__((ext_vector_type(2)));
typedef float v8f __attribute__((ext_vector_type(8)));

// ---------------------------------------------------------------------------
// Workspace zeroing (agg + cnt). Grid-stride.
// ---------------------------------------------------------------------------
__global__ void zero_ws_kernel(float* __restrict__ ws, size_t n) {
  size_t i = (size_t)blockIdx.x * blockDim.x + threadIdx.x;
  size_t stride = (size_t)gridDim.x * blockDim.x;
  for (; i < n; i += stride) ws[i] = 0.0f;
}

// ---------------------------------------------------------------------------
// Edge scatter: one wave (32 lanes) per edge. Each lane moves 4 floats of the
// 128-float source row via float4 gather + 4 hardware f32 atomics.
// ---------------------------------------------------------------------------
__global__ void edge_scatter_kernel(const float* __restrict__ xsrc,
                                    const int* __restrict__ es,
                                    const int* __restrict__ ed,
                                    float* __restrict__ agg,
                                    float* __restrict__ cnt,
                                    int E) {
  int gid = blockIdx.x * blockDim.x + threadIdx.x;
  int e = gid >> 5;
  int lane = gid & 31;
  if (e >= E) return;
  int s = es[e];
  int d = ed[e];
  const float4 v = *((const float4*)(xsrc + (size_t)s * H) + lane);
  float* dst = agg + (size_t)d * H + lane * 4;
  __hip_atomic_fetch_add(dst + 0, v.x, __ATOMIC_RELAXED, __HIP_MEMORY_SCOPE_AGENT);
  __hip_atomic_fetch_add(dst + 1, v.y, __ATOMIC_RELAXED, __HIP_MEMORY_SCOPE_AGENT);
  __hip_atomic_fetch_add(dst + 2, v.z, __ATOMIC_RELAXED, __HIP_MEMORY_SCOPE_AGENT);
  __hip_atomic_fetch_add(dst + 3, v.w, __ATOMIC_RELAXED, __HIP_MEMORY_SCOPE_AGENT);
  if (lane == 0)
    __hip_atomic_fetch_add(cnt + d, 1.0f, __ATOMIC_RELAXED, __HIP_MEMORY_SCOPE_AGENT);
}

// ---------------------------------------------------------------------------
// One 16x16 output tile accumulation: D += A(16xK) * B^T, B stored row-major
// as W[n][k] (n = output column). Uses V_WMMA_F32_16X16X4_F32.
// A fragment per lane: {A[l15][k+2h], A[l15][k+2h+1]}  (h = lane>>4)
// B fragment per lane: {W[n0+l15][k+2h], W[n0+l15][k+2h+1]}
// ---------------------------------------------------------------------------
template <int K>
__device__ __forceinline__ v8f wmma_accum(const float* __restrict__ A,
                                          const float* __restrict__ B,
                                          v8f acc, int l15, int half) {
  const float* arow = A + (size_t)l15 * K + 2 * half;
  const float* brow = B + (size_t)l15 * K + 2 * half;
#pragma unroll 4
  for (int k = 0; k < K; k += 4) {
    v2f a = *(const v2f*)(arow + k);
    v2f b = *(const v2f*)(brow + k);
    acc = __builtin_amdgcn_wmma_f32_16x16x4_f32(false, a, false, b, (short)0,
                                                acc, false, false);
  }
  return acc;
}

// ---------------------------------------------------------------------------
// Fused per-tile pipeline: mean -> {o0,o1,o2} -> relu(comb@W1^T+b1) -> @W2^T+b2
// Block = 128 threads (4 waves), one 16-user row tile per block.
// ---------------------------------------------------------------------------
__global__ __launch_bounds__(128) void fused_gemm_kernel(
    const float* __restrict__ xu,    // [NUSER][H]
    const float* __restrict__ agg,   // [3][NUSER][H]
    const float* __restrict__ cnt,   // [3][NUSER]
    const float* __restrict__ Wl,    // [3][H][H]
    const float* __restrict__ bl,    // [3][H]
    const float* __restrict__ Wr,    // [3][H][H]
    const float* __restrict__ W1,    // [H][3H]
    const float* __restrict__ b1,    // [H]
    const float* __restrict__ W2,    // [H][H]
    const float* __restrict__ b2,    // [H]
    float* __restrict__ out)         // [NUSER][H]
{
  __shared__ float meanL[3][16][H];
  __shared__ float xuL[16][H];
  __shared__ float combL[16][3 * H];
  __shared__ float hL[16][H];

  const int tid = threadIdx.x;
  const int lane = tid & 31;
  const int wave = tid >> 5;
  const int l15 = lane & 15;
  const int half = lane >> 4;
  const size_t m0 = (size_t)blockIdx.x * 16;

  // Phase 1: stage x_user tile and per-relation mean tiles into LDS.
  for (int i = tid; i < 16 * (H / 4); i += 128) {
    int row = i >> 5;             // 32 float4 chunks per 128-float row
    int c4 = (i & 31) << 2;
    *(float4*)&xuL[row][c4] = *(const float4*)(xu + (m0 + row) * H + c4);
#pragma unroll
    for (int r = 0; r < 3; ++r) {
      float c = cnt[(size_t)r * NUSER + m0 + row];
      float rc = 1.0f / fmaxf(c, 1.0f);
      float4 a = *(const float4*)(agg + ((size_t)r * NUSER + m0 + row) * H + c4);
      a.x *= rc; a.y *= rc; a.z *= rc; a.w *= rc;
      *(float4*)&meanL[r][row][c4] = a;
    }
  }
  __syncthreads();

  // Phase 2: o_r = mean_r @ Wl[r]^T + bl[r] + xu @ Wr[r]^T  -> combL.
  // 24 (relation, n-tile) units split across 4 waves.
  for (int u = wave; u < 24; u += 4) {
    int r = u >> 3;
    int nt = u & 7;
    int n0 = nt << 4;
    float bias = bl[r * H + n0 + l15];
    v8f acc;
#pragma unroll
    for (int v = 0; v < 8; ++v) acc[v] = bias;
    acc = wmma_accum<H>(&meanL[r][0][0], Wl + ((size_t)r * H + n0) * H, acc, l15, half);
    acc = wmma_accum<H>(&xuL[0][0],      Wr + ((size_t)r * H + n0) * H, acc, l15, half);
#pragma unroll
    for (int v = 0; v < 8; ++v)
      combL[v + 8 * half][r * H + n0 + l15] = acc[v];
  }
  __syncthreads();

  // Phase 3: h = relu(comb @ W1^T + b1)
  for (int nt = wave; nt < 8; nt += 4) {
    int n0 = nt << 4;
    float bias = b1[n0 + l15];
    v8f acc;
#pragma unroll
    for (int v = 0; v < 8; ++v) acc[v] = bias;
    acc = wmma_accum<3 * H>(&combL[0][0], W1 + (size_t)n0 * (3 * H), acc, l15, half);
#pragma unroll
    for (int v = 0; v < 8; ++v)
      hL[v + 8 * half][n0 + l15] = fmaxf(acc[v], 0.0f);
  }
  __syncthreads();

  // Phase 4: out = h @ W2^T + b2
  for (int nt = wave; nt < 8; nt += 4) {
    int n0 = nt << 4;
    float bias = b2[n0 + l15];
    v8f acc;
#pragma unroll
    for (int v = 0; v < 8; ++v) acc[v] = bias;
    acc = wmma_accum<H>(&hL[0][0], W2 + (size_t)n0 * H, acc, l15, half);
#pragma unroll
    for (int v = 0; v < 8; ++v)
      out[(m0 + v + 8 * half) * H + n0 + l15] = acc[v];
  }
}

// ---------------------------------------------------------------------------
extern "C" void kernel_launch(void* const* d_in, const int* in_sizes, int n_in,
                              void* d_out, int out_size, void* d_ws, size_t ws_size,
                              hipStream_t stream) {
  const float* x_user   = (const float*)d_in[0];
  const float* x_txn    = (const float*)d_in[1];
  const float* x_device = (const float*)d_in[2];
  const int* es0 = (const int*)d_in[3];
  const int* ed0 = (const int*)d_in[4];
  const int* es1 = (const int*)d_in[5];
  const int* ed1 = (const int*)d_in[6];
  const int* es2 = (const int*)d_in[7];
  const int* ed2 = (const int*)d_in[8];
  const float* Wl = (const float*)d_in[9];
  const float* bl = (const float*)d_in[10];
  const float* Wr = (const float*)d_in[11];
  const float* W1 = (const float*)d_in[12];
  const float* b1 = (const float*)d_in[13];
  const float* W2 = (const float*)d_in[14];
  const float* b2 = (const float*)d_in[15];
  float* out = (float*)d_out;
  const int E = in_sizes[3];

  float* agg = (float*)d_ws;                        // [3][NUSER][H]
  float* cnt = agg + (size_t)3 * NUSER * H;         // [3][NUSER]

  const size_t nz = (size_t)3 * NUSER * H + (size_t)3 * NUSER;
  zero_ws_kernel<<<2048, 256, 0, stream>>>(agg, nz);

  const int sblocks = (E * 32 + 255) / 256;
  edge_scatter_kernel<<<sblocks, 256, 0, stream>>>(x_txn, es0, ed0,
                                                   agg, cnt, E);
  edge_scatter_kernel<<<sblocks, 256, 0, stream>>>(x_device, es1, ed1,
                                                   agg + (size_t)NUSER * H,
                                                   cnt + NUSER, E);
  edge_scatter_kernel<<<sblocks, 256, 0, stream>>>(x_user, es2, ed2,
                                                   agg + (size_t)2 * NUSER * H,
                                                   cnt + 2 * NUSER, E);

  fused_gemm_kernel<<<NUSER / 16, 128, 0, stream>>>(
      x_user, agg, cnt, Wl, bl, Wr, W1, b1, W2, b2, out);
}